// MLS_21002390077701
// MI455X (gfx1250) — compile-verified
//
#include <hip/hip_runtime.h>
#include <math.h>

#define HH 128
#define WW 128
#define NN 68        // landmarks; 68 % 4 == 0 -> exact K loop
#define CC 256
#define BATCH 8
#define EPSV 1e-8f
#define HW (HH * WW)

typedef __attribute__((ext_vector_type(2))) float v2f;
typedef __attribute__((ext_vector_type(8))) float v8f;

// ---------------------------------------------------------------------------
// Kernel 1: MLS grid via WMMA.
//   S[m,k] = sum_n w[m,n] * G[n,k]   (M=16384, K=68, Ncols=16 with 12 used)
//   One wave = 16 pixels x 16 columns, 17x v_wmma_f32_16x16x4_f32.
// Block = 256 threads (8 waves, 128 pixels). Grid = 128 blocks.
// ---------------------------------------------------------------------------
__global__ __launch_bounds__(256) void mls_grid_kernel(
    const float* __restrict__ Ld, const float* __restrict__ Lg,
    float* __restrict__ grid) {
  __shared__ float ldx_s[NN];
  __shared__ float ldy_s[NN];
  __shared__ float g_s[NN * 16];
  __shared__ float sums_s[128 * 16];

  const int tid = threadIdx.x;

  // Build G (pixel-independent) and cache landmarks in LDS.
  if (tid < NN) {
    const float x = Ld[tid], y = Ld[NN + tid];
    const float gx = Lg[tid], gy = Lg[NN + tid];
    ldx_s[tid] = x;
    ldy_s[tid] = y;
    float* g = &g_s[tid * 16];
    g[0] = x * x;  g[1] = x * y;  g[2] = x;   g[3] = y * y;
    g[4] = y;      g[5] = 1.0f;   g[6] = gx * x; g[7] = gx * y;
    g[8] = gx;     g[9] = gy * x; g[10] = gy * y; g[11] = gy;
    g[12] = 0.0f;  g[13] = 0.0f;  g[14] = 0.0f;  g[15] = 0.0f;
  }
  __syncthreads();

  const int lane  = tid & 31;
  const int wave  = tid >> 5;
  const int col   = lane & 15;   // A: row M / C: column N
  const int khalf = lane >> 4;   // K pair select (K = 2*khalf + vgpr)

  // Pixel handled by this lane's A rows: m = i*W + j
  const int pix = (blockIdx.x * 8 + wave) * 16 + col;
  const int j = pix & (WW - 1);
  const int i = pix >> 7;
  const float px = (2.0f * (float)j + 1.0f) / (float)WW - 1.0f;
  const float py = (2.0f * (float)i + 1.0f) / (float)HH - 1.0f;

  v8f acc = {};
  for (int n = 0; n < NN; n += 4) {
    const int n0 = n + 2 * khalf;
    const float dx0 = ldx_s[n0] - px,     dy0 = ldy_s[n0] - py;
    const float dx1 = ldx_s[n0 + 1] - px, dy1 = ldy_s[n0 + 1] - py;
    v2f a, bm;
    a.x = 1.0f / (dx0 * dx0 + dy0 * dy0 + EPSV);
    a.y = 1.0f / (dx1 * dx1 + dy1 * dy1 + EPSV);
    bm.x = g_s[n0 * 16 + col];
    bm.y = g_s[(n0 + 1) * 16 + col];
    acc = __builtin_amdgcn_wmma_f32_16x16x4_f32(
        /*neg_a=*/false, a, /*neg_b=*/false, bm,
        /*c_mod=*/(short)0, acc, /*reuse_a=*/false, /*reuse_b=*/false);
  }

  // Scatter C fragment: lane holds column `col` for rows r (+8 if khalf).
  {
    float* dst = &sums_s[wave * 16 * 16];
#pragma unroll
    for (int r = 0; r < 8; ++r) dst[(khalf * 8 + r) * 16 + col] = acc[r];
  }
  __syncthreads();

  // One thread per pixel: symmetric 3x3 solve A s = hP, then pt = F s.
  if (tid < 128) {
    const float* s = &sums_s[tid * 16];
    const float a = s[0], b = s[1], c = s[2], d = s[3], e = s[4], f = s[5];
    const int p = blockIdx.x * 128 + tid;
    const int jj = p & (WW - 1);
    const int ii = p >> 7;
    const float X = (2.0f * (float)jj + 1.0f) / (float)WW - 1.0f;
    const float Y = (2.0f * (float)ii + 1.0f) / (float)HH - 1.0f;

    const float adj00 = d * f - e * e;
    const float adj01 = c * e - b * f;
    const float adj02 = b * e - c * d;
    const float adj11 = a * f - c * c;
    const float adj12 = b * c - a * e;
    const float adj22 = a * d - b * b;
    const float det = a * adj00 + b * adj01 + c * adj02;
    const float inv = 1.0f / det;
    const float s0 = (adj00 * X + adj01 * Y + adj02) * inv;
    const float s1 = (adj01 * X + adj11 * Y + adj12) * inv;
    const float s2 = (adj02 * X + adj12 * Y + adj22) * inv;
    const float ptx = s[6] * s0 + s[7] * s1 + s[8] * s2;
    const float pty = s[9] * s0 + s[10] * s1 + s[11] * s2;

    // Normalized -> image coords (align_corners=False).
    grid[2 * p + 0] = (ptx + 1.0f) * (0.5f * (float)WW) - 0.5f;
    grid[2 * p + 1] = (pty + 1.0f) * (0.5f * (float)HH) - 0.5f;
  }
}

// ---------------------------------------------------------------------------
// Kernel 2: bilinear gather. Grid shared across all (b,c): amortize weight
// math over the batch loop. Grid = (HW/256, C), block = 256.
// ---------------------------------------------------------------------------
__global__ __launch_bounds__(256) void mls_sample_kernel(
    const float* __restrict__ img, const float* __restrict__ grid,
    float* __restrict__ out) {
  const int p = blockIdx.x * 256 + threadIdx.x;  // pixel in [0, HW)
  const int c = blockIdx.y;

  const float x = grid[2 * p + 0];
  const float y = grid[2 * p + 1];
  const float x0f = floorf(x), y0f = floorf(y);
  const float wx1 = x - x0f, wx0 = 1.0f - wx1;
  const float wy1 = y - y0f, wy0 = 1.0f - wy1;
  const int x0 = (int)x0f, y0 = (int)y0f;
  const int x1 = x0 + 1, y1 = y0 + 1;

  const float vx0 = (x0 >= 0 && x0 < WW) ? 1.0f : 0.0f;
  const float vx1 = (x1 >= 0 && x1 < WW) ? 1.0f : 0.0f;
  const float vy0 = (y0 >= 0 && y0 < HH) ? 1.0f : 0.0f;
  const float vy1 = (y1 >= 0 && y1 < HH) ? 1.0f : 0.0f;

  const float w00 = wy0 * wx0 * vy0 * vx0;
  const float w01 = wy0 * wx1 * vy0 * vx1;
  const float w10 = wy1 * wx0 * vy1 * vx0;
  const float w11 = wy1 * wx1 * vy1 * vx1;

  const int x0c = min(max(x0, 0), WW - 1);
  const int x1c = min(max(x1, 0), WW - 1);
  const int y0c = min(max(y0, 0), HH - 1);
  const int y1c = min(max(y1, 0), HH - 1);
  const int i00 = y0c * WW + x0c;
  const int i01 = y0c * WW + x1c;
  const int i10 = y1c * WW + x0c;
  const int i11 = y1c * WW + x1c;

  const size_t chan_base = (size_t)c * HW;
  const size_t plane_stride = (size_t)CC * HW;

#pragma unroll
  for (int b = 0; b < BATCH; ++b) {
    const float* pl = img + chan_base + (size_t)b * plane_stride;
    const float v = pl[i00] * w00 + pl[i01] * w01 + pl[i10] * w10 + pl[i11] * w11;
    __builtin_nontemporal_store(v, out + chan_base + (size_t)b * plane_stride + p);
  }
}

extern "C" void kernel_launch(void* const* d_in, const int* in_sizes, int n_in,
                              void* d_out, int out_size, void* d_ws, size_t ws_size,
                              hipStream_t stream) {
  const float* fea_g = (const float*)d_in[0];  // [8,256,128,128]
  const float* L_d   = (const float*)d_in[1];  // [2,68]
  const float* L_g   = (const float*)d_in[2];  // [2,68]
  float* out  = (float*)d_out;                 // [8,256,128,128]
  float* grid = (float*)d_ws;                  // [16384,2] sample coords (128 KB)

  mls_grid_kernel<<<dim3(HW / 128), dim3(256), 0, stream>>>(L_d, L_g, grid);
  mls_sample_kernel<<<dim3(HW / 256, CC), dim3(256), 0, stream>>>(fea_g, grid, out);
}